// PowerMean_23046794510673
// MI455X (gfx1250) — compile-verified
//
#include <hip/hip_runtime.h>
#include <math.h>

// PowerMean(p = {1, 2, inf}) over masked sequence axis.
// x: [B, S=512, D=300] f32, sent_len: [B] i32  ->  out: [B, 900] f32
//   out[b, 0:300]   = sum(x[b, :len]) / len
//   out[b, 300:600] = sqrt(sum(x[b, :len]^2) / len)
//   out[b, 600:900] = max(x[b, :len])
//
// Bandwidth-bound (~0.6-1.3 GB read @ 23.3 TB/s). Strategy: gfx1250 async
// global->LDS (ASYNCcnt) 4-deep tile pipeline, VALU reduction from LDS.

#define D_DIM     300
#define S_DIM     512
#define TILE_ROWS 8
#define TILE_FLOATS (TILE_ROWS * D_DIM)   // 2400 floats = 9600 B per tile
#define TILE_CHUNKS (TILE_FLOATS / 4)     // 600 x 16B chunks
#define QDEPTH    4                        // tiles in flight
#define BLOCK     256
#define NEG_BIG   (-1e30f)

typedef int v4i __attribute__((vector_size(16)));
typedef __attribute__((address_space(1))) v4i as1_v4i;
typedef __attribute__((address_space(3))) v4i as3_v4i;
typedef __attribute__((address_space(3))) void as3_void_t;

// ---- gfx1250 async global->LDS copy (16B per lane), ASYNCcnt-tracked.
// cpol=1 -> TH=NT on the global read: one-touch stream, don't thrash L2.
__device__ __forceinline__ void async_copy_b128(const void* g, void* l) {
#if __has_builtin(__builtin_amdgcn_global_load_async_to_lds_b128)
  __builtin_amdgcn_global_load_async_to_lds_b128(
      (as1_v4i*)g, (as3_v4i*)l, /*imm offset*/ 0, /*cpol: TH_NT*/ 1);
#else
  unsigned lds_off = (unsigned)(unsigned long long)(as3_void_t*)l;
  asm volatile("global_load_async_to_lds_b128 %0, %1, off th:TH_LOAD_NT"
               :: "v"(lds_off), "v"((unsigned long long)g)
               : "memory");
#endif
}

#if __has_builtin(__builtin_amdgcn_s_wait_asynccnt)
#define WAIT_ASYNC(N) __builtin_amdgcn_s_wait_asynccnt(N)
#else
#define WAIT_ASYNC(N) asm volatile("s_wait_asynccnt %0" :: "n"(N) : "memory")
#endif

// Issue one tile (8 rows x 300 f32 = 600 x 16B chunks) as exactly 3 async
// b128 ops per thread. Chunk index is clamped (not predicated) so EXEC stays
// full and every wave's ASYNCcnt increases by exactly 3 per tile, making the
// wait constants below exact.
__device__ __forceinline__ void issue_tile(const char* gbase, char* lbase,
                                           int tid) {
#pragma unroll
  for (int i = 0; i < 3; ++i) {
    int c = tid + (i << 8);                       // tid, tid+256, tid+512
    c = (c < TILE_CHUNKS) ? c : (TILE_CHUNKS - 1);
    async_copy_b128(gbase + (size_t)c * 16, lbase + (size_t)c * 16);
  }
}

__global__ __launch_bounds__(BLOCK)
void powermean_kernel(const float* __restrict__ x,
                      const int* __restrict__ sent_len,
                      float* __restrict__ out) {
  __shared__ __align__(16) float smem[QDEPTH * TILE_FLOATS];  // 38400 B

  const int b   = blockIdx.x;
  const int tid = threadIdx.x;

  int len = sent_len[b];
  len = (len < 1) ? 1 : ((len > S_DIM) ? S_DIM : len);
  const int ntiles = (len + TILE_ROWS - 1) / TILE_ROWS;  // tiles stay within S

  const float* xb = x + (size_t)b * S_DIM * D_DIM;

  // Prime the pipeline: tiles 0 .. min(QDEPTH-1, ntiles)-1.
#pragma unroll
  for (int q = 0; q < QDEPTH - 1; ++q) {
    if (q < ntiles) {
      issue_tile((const char*)(xb + (size_t)q * TILE_FLOATS),
                 (char*)&smem[q * TILE_FLOATS], tid);
    }
  }

  const int d0 = tid;           // always < 300
  const int d1 = tid + BLOCK;   // valid for tid < 44
  float sum0 = 0.f, sq0 = 0.f, mx0 = NEG_BIG;
  float sum1 = 0.f, sq1 = 0.f, mx1 = NEG_BIG;

  for (int t = 0; t < ntiles; ++t) {
    const int nxt = t + (QDEPTH - 1);
    if (nxt < ntiles) {
      // Refill: overwrites the buffer last read at iteration t-1 (protected
      // by that iteration's trailing barrier).
      issue_tile((const char*)(xb + (size_t)nxt * TILE_FLOATS),
                 (char*)&smem[(nxt % QDEPTH) * TILE_FLOATS], tid);
    }
    // Async ops younger than tile t: 3 * min(QDEPTH-1, ntiles-1-t).
    const int rem = ntiles - 1 - t;
    if (rem >= 3)      WAIT_ASYNC(9);
    else if (rem == 2) WAIT_ASYNC(6);
    else if (rem == 1) WAIT_ASYNC(3);
    else               WAIT_ASYNC(0);
    __syncthreads();  // all waves' tile-t data visible in LDS

    const float* cur = &smem[(t % QDEPTH) * TILE_FLOATS];
    const int rend = min(TILE_ROWS, len - t * TILE_ROWS);
    for (int r = 0; r < rend; ++r) {
      const float* row = cur + r * D_DIM;  // lane-consecutive -> conflict-free
      float v0 = row[d0];
      sum0 += v0;
      sq0 = fmaf(v0, v0, sq0);
      mx0 = fmaxf(mx0, v0);
      if (d1 < D_DIM) {
        float v1 = row[d1];
        sum1 += v1;
        sq1 = fmaf(v1, v1, sq1);
        mx1 = fmaxf(mx1, v1);
      }
    }
    __syncthreads();  // done reading: buffer reusable for tile t+QDEPTH
  }

  const float inv = 1.0f / (float)len;
  float* ob = out + (size_t)b * (3 * D_DIM);
  ob[d0]             = sum0 * inv;
  ob[D_DIM + d0]     = sqrtf(sq0 * inv);
  ob[2 * D_DIM + d0] = mx0;
  if (d1 < D_DIM) {
    ob[d1]             = sum1 * inv;
    ob[D_DIM + d1]     = sqrtf(sq1 * inv);
    ob[2 * D_DIM + d1] = mx1;
  }
}

extern "C" void kernel_launch(void* const* d_in, const int* in_sizes, int n_in,
                              void* d_out, int out_size, void* d_ws, size_t ws_size,
                              hipStream_t stream) {
  (void)n_in; (void)out_size; (void)d_ws; (void)ws_size;
  const float* x        = (const float*)d_in[0];
  const int*   sent_len = (const int*)d_in[1];
  float*       out      = (float*)d_out;
  const int B = in_sizes[1];  // sent_len has one entry per batch row
  if (B <= 0) return;
  powermean_kernel<<<dim3((unsigned)B), dim3(BLOCK), 0, stream>>>(x, sent_len, out);
}